// StochasticVolatility_15006615733135
// MI455X (gfx1250) — compile-verified
//
#include <hip/hip_runtime.h>
#include <math.h>

// Heston SV Euler-Maruyama scan for MI455X (gfx1250, wave32).
//
// The nonlinear v-recurrence contracts with factor (1 - kappa*dt) = 0.996 per
// step, so a WARMUP-step run from an arbitrary start (theta) converges onto
// the true trajectory to ~1e-10 relative. Chunk the 2^23-step series:
//   pass 1: per-chunk warm-up + exact v scan; store v[], and the chunk's
//           sum of log-increments d[t] = (mu - 0.5 v)dt + sqrt(v dt) z2[t]
//           accumulated in DOUBLE (log-S reaches ~-666; f32 running sums
//           would inject ~1% random-walk error).
//   pass 2: exclusive prefix over the 8192 chunk log-sums (double, trivial)
//   pass 3: per-chunk: S[i] = S0 * expf(prefix + running double sum of d)
// Serial depth drops 8.4M -> 7168. Per-step dependent chain is kept to
// v_mul -> v_sqrt_f32 (raw builtin) -> v_fma -> v_max; everything else
// (d, sums, expf) is off-chain. Working set (~130MB) is L2-resident (192MB).

#define V_MIN_F  0.001f
#define CHUNK    1024
#define WARMUP   6144

__device__ __forceinline__ float vstep(float v, float z1t,
                                       float theta, float kdt, float xi, float dt) {
  float sq = __builtin_amdgcn_sqrtf(v * dt);  // chain: v_mul -> v_sqrt_f32
  float a  = fmaf(kdt, theta - v, v);         // off-chain vs sqrt
  float vn = fmaf(xi * z1t, sq, a);
  return fmaxf(vn, V_MIN_F);
}

// ---------------------------------------------------------------- pass 1 ----
__global__ void heston_v_scan(const float* __restrict__ z1,
                              const float* __restrict__ z2,
                              const float* __restrict__ p_mu,
                              const float* __restrict__ p_theta,
                              const float* __restrict__ p_kappa,
                              const float* __restrict__ p_xi,
                              const float* __restrict__ p_v0,
                              const float* __restrict__ p_dt,
                              float*  __restrict__ v_out,       // length M+1
                              double* __restrict__ chunk_sum,   // length nchunk
                              int M, int nchunk)
{
  const int c = blockIdx.x * blockDim.x + threadIdx.x;
  if (c >= nchunk) return;
  const float mu = p_mu[0], theta = p_theta[0], kappa = p_kappa[0];
  const float xi = p_xi[0], v0 = p_v0[0], dt = p_dt[0];
  const float kdt = kappa * dt;
  const int cC = c * CHUNK;

  float v;
  if (c == 0) {
    v = v0;
    v_out[0] = v0;                 // exact boundary element
  } else {
    // warm-up: start at theta, contract onto the true trajectory
    v = theta;
    const int w = (cC < WARMUP) ? cC : WARMUP;   // both multiples of CHUNK
    for (int k = cC - w; k < cC; k += 4) {
      if ((k & 63) == 0) __builtin_prefetch(z1 + k + 256, 0, 0);  // global_prefetch_b8
      float4 q = *reinterpret_cast<const float4*>(z1 + k);        // global_load_b128
      v = vstep(v, q.x, theta, kdt, xi, dt);
      v = vstep(v, q.y, theta, kdt, xi, dt);
      v = vstep(v, q.z, theta, kdt, xi, dt);
      v = vstep(v, q.w, theta, kdt, xi, dt);
    }
  }

  int nst = M - cC; if (nst > CHUNK) nst = CHUNK;
  double sum = 0.0;

#define MAIN_STEP(Z1, Z2, OUTI)                                         \
  { float sq = __builtin_amdgcn_sqrtf(v * dt);                          \
    float d  = fmaf(sq, (Z2), (mu - 0.5f * v) * dt);  /* uses v_prev */ \
    sum += (double)d;                            /* off the v chain */  \
    v = fmaxf(fmaf(xi * (Z1), sq, fmaf(kdt, theta - v, v)), V_MIN_F);   \
    v_out[(OUTI)] = v; }

  int k = 0;
  for (; k + 4 <= nst; k += 4) {
    const int i = cC + k;
    if ((k & 63) == 0) {
      __builtin_prefetch(z1 + i + 256, 0, 0);
      __builtin_prefetch(z2 + i + 256, 0, 0);
    }
    float4 q1 = *reinterpret_cast<const float4*>(z1 + i);
    float4 q2 = *reinterpret_cast<const float4*>(z2 + i);
    MAIN_STEP(q1.x, q2.x, i + 1)
    MAIN_STEP(q1.y, q2.y, i + 2)
    MAIN_STEP(q1.z, q2.z, i + 3)
    MAIN_STEP(q1.w, q2.w, i + 4)
  }
  for (; k < nst; ++k) {           // tail of the final (short) chunk
    const int i = cC + k;
    float a1 = z1[i], a2 = z2[i];
    MAIN_STEP(a1, a2, i + 1)
  }
#undef MAIN_STEP
  chunk_sum[c] = sum;
}

// ---------------------------------------------------------------- pass 2 ----
__global__ void heston_prefix(const double* __restrict__ sums,
                              double* __restrict__ pre, int nchunk)
{
  if (blockIdx.x == 0 && threadIdx.x == 0) {
    double acc = 0.0;
    for (int i = 0; i < nchunk; ++i) { pre[i] = acc; acc += sums[i]; }
  }
}

// ---------------------------------------------------------------- pass 3 ----
__global__ void heston_s_scan(const float* __restrict__ z2,
                              const float* __restrict__ v_arr,   // v_out of pass 1
                              const float* __restrict__ p_mu,
                              const float* __restrict__ p_S0,
                              const float* __restrict__ p_dt,
                              const double* __restrict__ chunk_pre,
                              float* __restrict__ S_out,
                              int M, int nchunk)
{
  const int c = blockIdx.x * blockDim.x + threadIdx.x;
  if (c >= nchunk) return;
  const float mu = p_mu[0], S0 = p_S0[0], dt = p_dt[0];
  const int cC = c * CHUNK;
  if (c == 0) S_out[0] = S0;

  double acc = chunk_pre[c];
  int nst = M - cC; if (nst > CHUNK) nst = CHUNK;

#define S_STEP(VP, Z2, OUTI)                                            \
  { float sq = __builtin_amdgcn_sqrtf((VP) * dt);                       \
    float d  = fmaf(sq, (Z2), (mu - 0.5f * (VP)) * dt);                 \
    acc += (double)d;              /* chain: cvt + v_add_f64 only */    \
    S_out[(OUTI)] = S0 * expf((float)acc); }  /* expf off the chain */

  int k = 0;
  for (; k + 4 <= nst; k += 4) {
    const int i = cC + k;
    if ((k & 63) == 0) {
      __builtin_prefetch(v_arr + i + 256, 0, 0);
      __builtin_prefetch(z2 + i + 256, 0, 0);
    }
    float4 qv = *reinterpret_cast<const float4*>(v_arr + i);
    float4 q2 = *reinterpret_cast<const float4*>(z2 + i);
    S_STEP(qv.x, q2.x, i + 1)
    S_STEP(qv.y, q2.y, i + 2)
    S_STEP(qv.z, q2.z, i + 3)
    S_STEP(qv.w, q2.w, i + 4)
  }
  for (; k < nst; ++k) {
    const int i = cC + k;
    float vp = v_arr[i], a2 = z2[i];
    S_STEP(vp, a2, i + 1)
  }
#undef S_STEP
}

// ------------------------------------------------------------------ host ----
extern "C" void kernel_launch(void* const* d_in, const int* in_sizes, int n_in,
                              void* d_out, int out_size, void* d_ws, size_t ws_size,
                              hipStream_t stream) {
  (void)n_in; (void)out_size; (void)ws_size;
  const float* z1    = (const float*)d_in[0];
  const float* z2    = (const float*)d_in[1];
  const float* mu    = (const float*)d_in[2];
  const float* theta = (const float*)d_in[3];
  const float* kappa = (const float*)d_in[4];
  const float* xi    = (const float*)d_in[5];
  const float* S0    = (const float*)d_in[6];
  const float* v0    = (const float*)d_in[7];
  const float* dt    = (const float*)d_in[8];

  const int M = in_sizes[0];        // n_steps - 1 random draws
  const int N = M + 1;              // output series length
  const int nchunk = (M + CHUNK - 1) / CHUNK;

  float*  S_out = (float*)d_out;        // outputs concatenated: S then v
  float*  v_out = S_out + N;
  double* sums  = (double*)d_ws;        // [nchunk] chunk log-sums (f64)
  double* pre   = sums + nchunk;        // [nchunk] exclusive prefix (f64)

  const int tb = 256;                   // 8 waves (wave32) per block
  const int gb = (nchunk + tb - 1) / tb;

  heston_v_scan<<<gb, tb, 0, stream>>>(z1, z2, mu, theta, kappa, xi, v0, dt,
                                       v_out, sums, M, nchunk);
  heston_prefix<<<1, 1, 0, stream>>>(sums, pre, nchunk);
  heston_s_scan<<<gb, tb, 0, stream>>>(z2, v_out, mu, S0, dt, pre,
                                       S_out, M, nchunk);
}